// WindowSelfAttention2D_81097572483357
// MI455X (gfx1250) — compile-verified
//
#include <hip/hip_runtime.h>
#include <hip/hip_bf16.h>
#include <math.h>

// Problem constants (from reference)
#define BATCH   2
#define CH      128      // C == INNER
#define HDIM    96
#define WDIM    96
#define PIX     (HDIM * WDIM)   // 9216
#define HEADS   4
#define DHEAD   32
#define WIN     5
#define NPOS    (WIN * WIN)     // 25
#define PADR    2

typedef __attribute__((ext_vector_type(2))) float v2f;
typedef __attribute__((ext_vector_type(8))) float v8f;

#define MTILES  (CH / 16)         // 8
#define NT      4                 // N-tiles (16 cols each) per wave -> 16x64 strip
#define PTILES4 (PIX / (16 * NT)) // 144
#define GEMM_WAVES  (BATCH * MTILES * PTILES4)  // 2304 waves
#define GEMM_BLOCKS (GEMM_WAVES / 8)            // 288 blocks of 256 thr

// ---------------------------------------------------------------------------
// GEMM:  Y[b, m, p] = sum_k Wm[m, k] * X[b, k, p]
//   Wm row-major (128 x 128); X/Y: (BATCH, 128, PIX)
// One wave computes a 16x64 (m,p) strip: 4 accumulators, K=128 in steps of 4
// via v_wmma_f32_16x16x4_f32 (exact fp32). Per K-step: one b64 A load shared
// by 4 WMMAs; all B loads are immediate-offset b32 loads off one base ptr.
//
// Layouts (ISA 7.12.2, f32):
//  A (16x4): lane-half h, VGPR v -> K = v + 2h, M = lane&15   (contiguous pair)
//  B (4x16): lane-half h, VGPR v -> K = v + 2h, N = lane&15
//  C/D (16x16): VGPR j -> row j + 8h, col lane&15
// ---------------------------------------------------------------------------
__global__ __launch_bounds__(256) void gemm128_wmma(
    const float* __restrict__ Wm,
    const float* __restrict__ X,
    float* __restrict__ Y)
{
    const int wave = (blockIdx.x * blockDim.x + threadIdx.x) >> 5;
    const int lane = threadIdx.x & 31;
    const int half = lane >> 4;     // 0 or 1
    const int l    = lane & 15;

    const int pt = wave % PTILES4;          // which 64-wide strip
    const int r  = wave / PTILES4;
    const int mt = r % MTILES;
    const int b  = r / MTILES;

    const int m = mt * 16 + l;              // A row for this lane
    // Single base pointer for all B loads: everything else is an immediate
    // offset (max (127*PIX + 48)*4 B ~= 4.7 MB, inside the signed-24-bit range).
    const float* __restrict__ Xp = X + (size_t)b * CH * PIX + pt * (16 * NT) + l;
    const float* __restrict__ Wrow = Wm + m * CH;

    v8f acc0 = {}, acc1 = {}, acc2 = {}, acc3 = {};

    #pragma unroll
    for (int k0 = 0; k0 < CH; k0 += 4) {
        const int ka = k0 + 2 * half;       // even -> 8-byte aligned
        const v2f a = *(const v2f*)(Wrow + ka);

        v2f b0, b1, b2, b3;
        b0.x = Xp[ka * PIX +  0]; b0.y = Xp[(ka + 1) * PIX +  0];
        b1.x = Xp[ka * PIX + 16]; b1.y = Xp[(ka + 1) * PIX + 16];
        b2.x = Xp[ka * PIX + 32]; b2.y = Xp[(ka + 1) * PIX + 32];
        b3.x = Xp[ka * PIX + 48]; b3.y = Xp[(ka + 1) * PIX + 48];

        acc0 = __builtin_amdgcn_wmma_f32_16x16x4_f32(false, a, false, b0,
                                                     (short)0, acc0, false, false);
        acc1 = __builtin_amdgcn_wmma_f32_16x16x4_f32(false, a, false, b1,
                                                     (short)0, acc1, false, false);
        acc2 = __builtin_amdgcn_wmma_f32_16x16x4_f32(false, a, false, b2,
                                                     (short)0, acc2, false, false);
        acc3 = __builtin_amdgcn_wmma_f32_16x16x4_f32(false, a, false, b3,
                                                     (short)0, acc3, false, false);
    }

    float* __restrict__ Yb = Y + (size_t)b * CH * PIX
                               + (size_t)(mt * 16) * PIX + pt * (16 * NT) + l;
    #pragma unroll
    for (int j = 0; j < 8; ++j) {
        const size_t row = (size_t)(j + 8 * half) * PIX;
        Yb[row +  0] = acc0[j];
        Yb[row + 16] = acc1[j];
        Yb[row + 32] = acc2[j];
        Yb[row + 48] = acc3[j];
    }
}

// ---------------------------------------------------------------------------
// Windowed attention: one thread per (b, head, pixel).
// Scores: q . k_shift over d=32 for 25 offsets; OOB -> logit 0 (zero-padded k).
// Softmax over 25, then weighted sum of v (OOB v contributes 0).
// ---------------------------------------------------------------------------
__global__ __launch_bounds__(256) void win_attn(
    const float* __restrict__ Q,
    const float* __restrict__ Kf,
    const float* __restrict__ Vf,
    float* __restrict__ O)
{
    const int idx = blockIdx.x * blockDim.x + threadIdx.x; // < BATCH*HEADS*PIX
    const int p = idx % PIX;
    const int t = idx / PIX;
    const int h = t % HEADS;
    const int b = t / HEADS;

    const int py = p / WDIM;
    const int px = p % WDIM;

    const size_t base = ((size_t)b * CH + h * DHEAD) * PIX;
    const float* __restrict__ Qb = Q  + base;
    const float* __restrict__ Kb = Kf + base;
    const float* __restrict__ Vb = Vf + base;

    float q[DHEAD];
    #pragma unroll
    for (int d = 0; d < DHEAD; ++d) q[d] = Qb[(size_t)d * PIX + p];

    const float scale = 0.1767766952966369f; // 1/sqrt(32)
    float s[NPOS];

    #pragma unroll
    for (int di = 0; di < WIN; ++di) {
        #pragma unroll
        for (int dj = 0; dj < WIN; ++dj) {
            const int yy = py + di - PADR;
            const int xx = px + dj - PADR;
            float acc = 0.0f;
            if (yy >= 0 && yy < HDIM && xx >= 0 && xx < WDIM) {
                const int pp = yy * WDIM + xx;
                #pragma unroll
                for (int d = 0; d < DHEAD; ++d)
                    acc = fmaf(q[d], Kb[(size_t)d * PIX + pp], acc);
                acc *= scale;
            }
            s[di * WIN + dj] = acc; // zero-padded k => logit exactly 0 OOB
        }
    }

    float mx = s[0];
    #pragma unroll
    for (int k = 1; k < NPOS; ++k) mx = fmaxf(mx, s[k]);
    float sum = 0.0f;
    #pragma unroll
    for (int k = 0; k < NPOS; ++k) { s[k] = expf(s[k] - mx); sum += s[k]; }
    const float inv = 1.0f / sum;

    float out[DHEAD];
    #pragma unroll
    for (int d = 0; d < DHEAD; ++d) out[d] = 0.0f;

    #pragma unroll
    for (int di = 0; di < WIN; ++di) {
        #pragma unroll
        for (int dj = 0; dj < WIN; ++dj) {
            const int yy = py + di - PADR;
            const int xx = px + dj - PADR;
            if (yy >= 0 && yy < HDIM && xx >= 0 && xx < WDIM) {
                const int pp = yy * WDIM + xx;
                const float w = s[di * WIN + dj] * inv;
                #pragma unroll
                for (int d = 0; d < DHEAD; ++d)
                    out[d] = fmaf(w, Vb[(size_t)d * PIX + pp], out[d]);
            }
        }
    }

    #pragma unroll
    for (int d = 0; d < DHEAD; ++d)
        O[base + (size_t)d * PIX + p] = out[d];
}

// ---------------------------------------------------------------------------
// Launch: q/k/v GEMMs -> attention -> projection GEMM.
// Workspace: Q, K, V, AO each BATCH*CH*PIX floats (4 x 9.44 MB).
// ---------------------------------------------------------------------------
extern "C" void kernel_launch(void* const* d_in, const int* in_sizes, int n_in,
                              void* d_out, int out_size, void* d_ws, size_t ws_size,
                              hipStream_t stream) {
    (void)in_sizes; (void)n_in; (void)out_size; (void)ws_size;

    const float* x     = (const float*)d_in[0];
    const float* wq    = (const float*)d_in[1];
    const float* wk    = (const float*)d_in[2];
    const float* wv    = (const float*)d_in[3];
    const float* wproj = (const float*)d_in[4];
    float* out = (float*)d_out;

    const size_t planeN = (size_t)BATCH * CH * PIX; // 2359296 floats
    float* Qbuf = (float*)d_ws;
    float* Kbuf = Qbuf + planeN;
    float* Vbuf = Kbuf + planeN;
    float* AO   = Vbuf + planeN;

    gemm128_wmma<<<GEMM_BLOCKS, 256, 0, stream>>>(wq, x, Qbuf);
    gemm128_wmma<<<GEMM_BLOCKS, 256, 0, stream>>>(wk, x, Kbuf);
    gemm128_wmma<<<GEMM_BLOCKS, 256, 0, stream>>>(wv, x, Vbuf);

    const int attn_threads = BATCH * HEADS * PIX;   // 73728
    win_attn<<<attn_threads / 256, 256, 0, stream>>>(Qbuf, Kbuf, Vbuf, AO);

    gemm128_wmma<<<GEMM_BLOCKS, 256, 0, stream>>>(wproj, AO, out);
}